// LearnableEncoding_7017976562283
// MI455X (gfx1250) — compile-verified
//
#include <hip/hip_runtime.h>
#include <cstdint>
#include <cstddef>

// -------------------------------------------------------------------------
// LearnableEncoding: 8-level multi-resolution bilinear feature gather.
//   lat_idx = round_ne((90 - lat)/0.1) in [0,1800]
//   lon_idx = round_ne(lon/0.1)        in [0,3599]
//   per level i: bilinear(align_corners=False) sample of grid_i upsampled
//   to 1801x3600, evaluated only at (lat_idx, lon_idx).
// Levels 4..7 grids (~136 KB) staged in LDS via CDNA5 async global->LDS
// (b128 bulk + b32 tail, ASYNCcnt-tracked); levels 0..3 gathered from
// global (grids total ~35 MB -> fully L2-resident). Output/coords streamed
// nontemporally so the near caches stay dedicated to the gather-hot grids.
// -------------------------------------------------------------------------

// Grid shapes: H_i = ceil(1801/2^i), W_i = ceil(3600/2^i)
// i:      0     1    2    3    4    5   6   7
// H:   1801   901  451  226  113   57  29  15
// W:   3600  1800  900  450  225  113  57  29

constexpr int SM4 = 113 * 225;  // 25425 floats
constexpr int SM5 = 57 * 113;   //  6441
constexpr int SM6 = 29 * 57;    //  1653
constexpr int SM7 = 15 * 29;    //   435

// 16-byte-aligned LDS sub-bases (pad each level to a multiple of 4 floats)
constexpr int O4 = 0;
constexpr int O5 = (O4 + SM4 + 3) & ~3;  // 25428
constexpr int O6 = (O5 + SM5 + 3) & ~3;  // 31872
constexpr int O7 = (O6 + SM6 + 3) & ~3;  // 33528
constexpr int SMTOT = O7 + SM7;          // 33963 floats = 135852 B LDS

constexpr int NTHREADS = 512;
constexpr int NBLOCKS  = 512;

// Native clang vectors (accepted by __builtin_nontemporal_load/store).
typedef float f32x2 __attribute__((ext_vector_type(2)));
typedef float f32x4 __attribute__((ext_vector_type(4)));

// 4-byte-aligned pair so the compiler may emit a single b64 load (unaligned
// global access is supported) while the code stays well-defined either way.
struct __attribute__((aligned(4))) FPair { float a, b; };

struct BL { int y0, y1, x0; float wy, wx; };

template <int HIN, int WIN>
__device__ __forceinline__ BL mkbl(int yi, int xi) {
  // scale constants exactly as the reference: Hin/H_out computed in double,
  // rounded to f32, then used in an f32 multiply.
  const float scy = (float)((double)HIN / 1801.0);
  const float scx = (float)((double)WIN / 3600.0);
  float sy = fmaxf(((float)yi + 0.5f) * scy - 0.5f, 0.0f);
  float sx = fmaxf(((float)xi + 0.5f) * scx - 0.5f, 0.0f);
  BL b;
  b.y0 = (int)sy;                    // floor: 0 <= sy < HIN
  b.wy = sy - (float)b.y0;
  b.y1 = min(b.y0 + 1, HIN - 1);
  int x0 = (int)sx;                  // floor
  // Fold the x1 = min(x0+1, W-1) edge clamp into (x0c, wx): when x0 == W-1
  // the reference result is exactly g[y, W-1]; with x0c = W-2 and wx = 1 the
  // pair-interpolation reproduces that identically.
  b.x0 = min(x0, WIN - 2);
  b.wx = fminf(sx - (float)b.x0, 1.0f);
  return b;
}

// Global-memory bilinear sample: two 8-byte row-pair gathers per level.
template <int HIN, int WIN>
__device__ __forceinline__ float sampleG(const float* __restrict__ g, int yi, int xi) {
  BL b = mkbl<HIN, WIN>(yi, xi);
  FPair r0 = *reinterpret_cast<const FPair*>(g + (size_t)b.y0 * WIN + b.x0);
  FPair r1 = *reinterpret_cast<const FPair*>(g + (size_t)b.y1 * WIN + b.x0);
  float top = r0.a * (1.0f - b.wx) + r0.b * b.wx;
  float bot = r1.a * (1.0f - b.wx) + r1.b * b.wx;
  return top * (1.0f - b.wy) + bot * b.wy;
}

// LDS bilinear sample (pointer derived from __shared__; inlined so address-
// space inference emits ds_load_2addr_b32 row pairs).
template <int HIN, int WIN>
__device__ __forceinline__ float sampleL(const float* s, int yi, int xi) {
  BL b = mkbl<HIN, WIN>(yi, xi);
  float v00 = s[b.y0 * WIN + b.x0];
  float v01 = s[b.y0 * WIN + b.x0 + 1];
  float v10 = s[b.y1 * WIN + b.x0];
  float v11 = s[b.y1 * WIN + b.x0 + 1];
  float top = v00 * (1.0f - b.wx) + v01 * b.wx;
  float bot = v10 * (1.0f - b.wx) + v11 * b.wx;
  return top * (1.0f - b.wy) + bot * b.wy;
}

// Cooperative async fill: CDNA5 global_load_async_to_lds (GVS mode:
// saddr = grid base SGPR pair, vaddr = per-lane byte offset, vdst = per-lane
// LDS byte offset). Tracked by ASYNCcnt; no VGPR round-trip. Bulk uses the
// 16 B/lane b128 form (both src and LDS sub-base are 16 B aligned); the
// <=3-element tail uses b32.
__device__ __forceinline__ void fill_async(float* dst_lds, const float* __restrict__ src,
                                           int n, unsigned tid, unsigned nthr) {
  unsigned base = (unsigned)(size_t)(void*)dst_lds;  // low 32 bits == LDS offset
  const unsigned n16 = (unsigned)n >> 2;             // # of 4-float chunks
  for (unsigned i = tid; i < n16; i += nthr) {
    unsigned lds  = base + i * 16u;
    unsigned goff = i * 16u;
    asm volatile("global_load_async_to_lds_b128 %0, %1, %2"
                 :: "v"(lds), "v"(goff), "s"(src)
                 : "memory");
  }
  for (unsigned i = n16 * 4u + tid; i < (unsigned)n; i += nthr) {
    unsigned lds  = base + i * 4u;
    unsigned goff = i * 4u;
    asm volatile("global_load_async_to_lds_b32 %0, %1, %2"
                 :: "v"(lds), "v"(goff), "s"(src)
                 : "memory");
  }
}

__global__ __launch_bounds__(NTHREADS)
void learnable_encoding_kernel(const float* __restrict__ x,
                               const float* __restrict__ g0,
                               const float* __restrict__ g1,
                               const float* __restrict__ g2,
                               const float* __restrict__ g3,
                               const float* __restrict__ g4,
                               const float* __restrict__ g5,
                               const float* __restrict__ g6,
                               const float* __restrict__ g7,
                               float* __restrict__ out,
                               int npts) {
  __shared__ float smem[SMTOT];

  const unsigned tid  = threadIdx.x;
  const unsigned nthr = blockDim.x;

  // Stage the four small grids (levels 4..7) into LDS.
  fill_async(smem + O4, g4, SM4, tid, nthr);
  fill_async(smem + O5, g5, SM5, tid, nthr);
  fill_async(smem + O6, g6, SM6, tid, nthr);
  fill_async(smem + O7, g7, SM7, tid, nthr);
  asm volatile("s_wait_asynccnt 0" ::: "memory");
  __syncthreads();

  const float* s4 = smem + O4;
  const float* s5 = smem + O5;
  const float* s6 = smem + O6;
  const float* s7 = smem + O7;

  const int stride = (int)(gridDim.x * blockDim.x);
  for (int n = (int)(blockIdx.x * blockDim.x + threadIdx.x); n < npts; n += stride) {
    const f32x2 p =
        __builtin_nontemporal_load(reinterpret_cast<const f32x2*>(x) + n);
    const float lat = p.x;
    const float lon = p.y;
    // coord -> index, round-half-to-even, matching jnp.round + /0.1f division
    const int yi = __float2int_rn((90.0f - lat) / 0.1f);
    const int xi = __float2int_rn(lon / 0.1f);

    float r0 = sampleG<1801, 3600>(g0, yi, xi);
    float r1 = sampleG< 901, 1800>(g1, yi, xi);
    float r2 = sampleG< 451,  900>(g2, yi, xi);
    float r3 = sampleG< 226,  450>(g3, yi, xi);
    float r4 = sampleL< 113,  225>(s4, yi, xi);
    float r5 = sampleL<  57,  113>(s5, yi, xi);
    float r6 = sampleL<  29,   57>(s6, yi, xi);
    float r7 = sampleL<  15,   29>(s7, yi, xi);

    // Streamed once -> nontemporal so grids keep the near caches.
    f32x4* o = reinterpret_cast<f32x4*>(out + (size_t)n * 8);
    f32x4 lo = {r0, r1, r2, r3};
    f32x4 hi = {r4, r5, r6, r7};
    __builtin_nontemporal_store(lo, o);
    __builtin_nontemporal_store(hi, o + 1);
  }
}

extern "C" void kernel_launch(void* const* d_in, const int* in_sizes, int n_in,
                              void* d_out, int out_size, void* d_ws, size_t ws_size,
                              hipStream_t stream) {
  (void)n_in; (void)out_size; (void)d_ws; (void)ws_size;
  const float* x  = (const float*)d_in[0];
  const float* g0 = (const float*)d_in[1];
  const float* g1 = (const float*)d_in[2];
  const float* g2 = (const float*)d_in[3];
  const float* g3 = (const float*)d_in[4];
  const float* g4 = (const float*)d_in[5];
  const float* g5 = (const float*)d_in[6];
  const float* g6 = (const float*)d_in[7];
  const float* g7 = (const float*)d_in[8];
  float* out = (float*)d_out;
  const int npts = in_sizes[0] / 2;

  learnable_encoding_kernel<<<NBLOCKS, NTHREADS, 0, stream>>>(
      x, g0, g1, g2, g3, g4, g5, g6, g7, out, npts);
}